// Prox_44530220925112
// MI455X (gfx1250) — compile-verified
//
#include <hip/hip_runtime.h>
#include <math.h>

#define B_   32
#define C_   4
#define L_   (512*512)         // 262144 pixels per batch
#define NROW (B_*C_)           // 128 selection rows
#define K_EN 2621u             // rank of 'en' (0-indexed, descending)
#define K_ST 259522u           // rank of 'st'
#define SEG  64                // segments per batch for histogram passes
#define PIX_PER_BLK (L_/SEG)   // 4096 pixels per block
#define TPB  256
#define ITERS (PIX_PER_BLK/TPB) // 16 tiles of 256 float4 per block

// ---- order-preserving float <-> u32 key ---------------------------------
__device__ __forceinline__ unsigned f2ord(float x) {
    unsigned b = __float_as_uint(x);
    return (b & 0x80000000u) ? ~b : (b | 0x80000000u);
}
__device__ __forceinline__ float ord2f(unsigned u) {
    unsigned b = (u & 0x80000000u) ? (u & 0x7FFFFFFFu) : ~u;
    return __uint_as_float(b);
}

// ---- gfx1250 async global->LDS copy (ASYNCcnt-tracked) -------------------
// Shared-aperture generic pointers carry the LDS byte address in the low 32
// bits (ISA 10.2).
__device__ __forceinline__ unsigned lds_addr_of(const void* p) {
    return (unsigned)(uintptr_t)p;
}
__device__ __forceinline__ void async_copy_b128(unsigned lds, const void* gaddr) {
    asm volatile("global_load_async_to_lds_b128 %0, %1, off"
                 :: "v"(lds), "v"((unsigned long long)(uintptr_t)gaddr)
                 : "memory");
}
__device__ __forceinline__ void wait_async_le1() {
    asm volatile("s_wait_asynccnt 1" ::: "memory");
}
__device__ __forceinline__ void wait_async_0() {
    asm volatile("s_wait_asynccnt 0" ::: "memory");
}

// ---- zero a scratch region ----------------------------------------------
__global__ void zero_ws(unsigned* __restrict__ p, int n) {
    int i = blockIdx.x * blockDim.x + threadIdx.x;
    if (i < n) p[i] = 0u;
}

// ---- level 0: histogram of top 11 bits, per (b,c) -----------------------
// grid (SEG, B_), 256 threads. LDS: 32 KB hist + 8 KB double-buffer stage.
__global__ void hist_l0(const float4* __restrict__ x, unsigned* __restrict__ gh) {
    __shared__ unsigned h[C_ * 2048];
    __shared__ float4 stage[2][TPB];
    const int t = threadIdx.x;
    for (int i = t; i < C_ * 2048; i += TPB) h[i] = 0u;
    __syncthreads();

    const int b = blockIdx.y, seg = blockIdx.x;
    const float4* base = x + (size_t)b * L_ + (size_t)seg * PIX_PER_BLK;
    const unsigned ldsA = lds_addr_of(&stage[0][t]);
    const unsigned ldsB = lds_addr_of(&stage[1][t]);

    async_copy_b128(ldsA, base + t);                       // prime tile 0
    for (int it = 0; it < ITERS; ++it) {
        if (it + 1 < ITERS) {
            async_copy_b128((it + 1) & 1 ? ldsB : ldsA, base + (it + 1) * TPB + t);
            wait_async_le1();                              // tile 'it' landed
        } else {
            wait_async_0();
        }
        float4 v = stage[it & 1][t];                       // ds_load_b128
        atomicAdd(&h[0 * 2048 + (f2ord(v.x) >> 21)], 1u);  // ds_add_u32
        atomicAdd(&h[1 * 2048 + (f2ord(v.y) >> 21)], 1u);
        atomicAdd(&h[2 * 2048 + (f2ord(v.z) >> 21)], 1u);
        atomicAdd(&h[3 * 2048 + (f2ord(v.w) >> 21)], 1u);
    }
    __syncthreads();

    unsigned* dst = gh + (size_t)b * C_ * 2048;   // layout [row = b*4+c][bin]
    for (int i = t; i < C_ * 2048; i += TPB) {
        unsigned v = h[i];
        if (v) atomicAdd(&dst[i], v);             // global_atomic_add_u32
    }
}

// ---- level 1: next 11 bits, conditioned on prefix, both ranks packed ----
__global__ void hist_l1(const float4* __restrict__ x, const uint2* __restrict__ state,
                        unsigned* __restrict__ gh) {
    __shared__ unsigned h[C_ * 2048];
    __shared__ float4 stage[2][TPB];
    const int t = threadIdx.x;
    for (int i = t; i < C_ * 2048; i += TPB) h[i] = 0u;

    const int b = blockIdx.y, seg = blockIdx.x;
    unsigned pfA[C_], pfB[C_];
    #pragma unroll
    for (int c = 0; c < C_; ++c) {
        pfA[c] = state[(b * C_ + c) * 2 + 0].x;   // en prefix (11 bits)
        pfB[c] = state[(b * C_ + c) * 2 + 1].x;   // st prefix
    }
    __syncthreads();

    const float4* base = x + (size_t)b * L_ + (size_t)seg * PIX_PER_BLK;
    const unsigned ldsA = lds_addr_of(&stage[0][t]);
    const unsigned ldsB = lds_addr_of(&stage[1][t]);

    async_copy_b128(ldsA, base + t);
    for (int it = 0; it < ITERS; ++it) {
        if (it + 1 < ITERS) {
            async_copy_b128((it + 1) & 1 ? ldsB : ldsA, base + (it + 1) * TPB + t);
            wait_async_le1();
        } else {
            wait_async_0();
        }
        float4 v = stage[it & 1][t];
        const float vv[4] = {v.x, v.y, v.z, v.w};
        #pragma unroll
        for (int c = 0; c < C_; ++c) {
            unsigned u   = f2ord(vv[c]);
            unsigned hi  = u >> 21;
            unsigned mid = (u >> 10) & 0x7FFu;
            unsigned add = (hi == pfA[c] ? 1u : 0u) + (hi == pfB[c] ? 0x10000u : 0u);
            if (add) atomicAdd(&h[c * 2048 + mid], add);   // counts <= 4096: no carry
        }
    }
    __syncthreads();

    unsigned* dst = gh + (size_t)b * C_ * 2 * 2048;   // [(row*2+rank)][bin]
    for (int i = t; i < C_ * 2048; i += TPB) {
        unsigned v = h[i];
        int c = i >> 11, bin = i & 0x7FF;
        unsigned lo = v & 0xFFFFu, hiC = v >> 16;
        if (lo)  atomicAdd(&dst[(c * 2 + 0) * 2048 + bin], lo);
        if (hiC) atomicAdd(&dst[(c * 2 + 1) * 2048 + bin], hiC);
    }
}

// ---- level 2: final 10 bits ---------------------------------------------
__global__ void hist_l2(const float4* __restrict__ x, const uint2* __restrict__ state,
                        unsigned* __restrict__ gh) {
    __shared__ unsigned h[C_ * 1024];
    __shared__ float4 stage[2][TPB];
    const int t = threadIdx.x;
    for (int i = t; i < C_ * 1024; i += TPB) h[i] = 0u;

    const int b = blockIdx.y, seg = blockIdx.x;
    unsigned pfA[C_], pfB[C_];
    #pragma unroll
    for (int c = 0; c < C_; ++c) {
        pfA[c] = state[(b * C_ + c) * 2 + 0].x;   // 22-bit prefixes
        pfB[c] = state[(b * C_ + c) * 2 + 1].x;
    }
    __syncthreads();

    const float4* base = x + (size_t)b * L_ + (size_t)seg * PIX_PER_BLK;
    const unsigned ldsA = lds_addr_of(&stage[0][t]);
    const unsigned ldsB = lds_addr_of(&stage[1][t]);

    async_copy_b128(ldsA, base + t);
    for (int it = 0; it < ITERS; ++it) {
        if (it + 1 < ITERS) {
            async_copy_b128((it + 1) & 1 ? ldsB : ldsA, base + (it + 1) * TPB + t);
            wait_async_le1();
        } else {
            wait_async_0();
        }
        float4 v = stage[it & 1][t];
        const float vv[4] = {v.x, v.y, v.z, v.w};
        #pragma unroll
        for (int c = 0; c < C_; ++c) {
            unsigned u   = f2ord(vv[c]);
            unsigned hi  = u >> 10;
            unsigned lo  = u & 0x3FFu;
            unsigned add = (hi == pfA[c] ? 1u : 0u) + (hi == pfB[c] ? 0x10000u : 0u);
            if (add) atomicAdd(&h[c * 1024 + lo], add);
        }
    }
    __syncthreads();

    unsigned* dst = gh + (size_t)b * C_ * 2 * 1024;   // [(row*2+rank)][bin]
    for (int i = t; i < C_ * 1024; i += TPB) {
        unsigned v = h[i];
        int c = i >> 10, bin = i & 0x3FF;
        unsigned lo = v & 0xFFFFu, hiC = v >> 16;
        if (lo)  atomicAdd(&dst[(c * 2 + 0) * 1024 + bin], lo);
        if (hiC) atomicAdd(&dst[(c * 2 + 1) * 1024 + bin], hiC);
    }
}

// ---- digit selection via parallel suffix-scan (top digit downward) ------
// grid = NROW*2 blocks (row, rank), 256 threads.
__global__ void select_lvl(const unsigned* __restrict__ gh, uint2* __restrict__ state,
                           int level) {
    const int row  = blockIdx.x >> 1;
    const int rank = blockIdx.x & 1;
    const int NB   = (level == 2) ? 1024 : 2048;
    const int chunk = NB / TPB;                 // 8 or 4

    const unsigned* hist;
    unsigned rem, pfx = 0;
    if (level == 0) {
        hist = gh + (size_t)row * 2048;
        rem  = rank ? K_ST : K_EN;
    } else {
        uint2 s = state[row * 2 + rank];
        pfx = s.x; rem = s.y;
        hist = gh + (size_t)(row * 2 + rank) * NB;
    }

    __shared__ unsigned bufA[TPB], bufB[TPB];
    __shared__ unsigned selDigit, selBase;
    const int t = threadIdx.x;

    // walk bins from the top (reversed index); local copy + chunk sum
    unsigned loc[8];
    unsigned s0 = 0;
    for (int j = 0; j < chunk; ++j) {
        int i = t * chunk + j;
        loc[j] = hist[NB - 1 - i];
        s0 += loc[j];
    }
    bufA[t] = s0;
    __syncthreads();

    // Hillis-Steele inclusive scan of chunk sums (double buffer)
    unsigned* in = bufA; unsigned* out = bufB;
    for (int off = 1; off < TPB; off <<= 1) {
        unsigned v = in[t];
        if (t >= off) v += in[t - off];
        out[t] = v;
        __syncthreads();
        unsigned* tmp = in; in = out; out = tmp;
    }
    unsigned run = in[t] - s0;   // count of elements with strictly greater digit

    for (int j = 0; j < chunk; ++j) {
        int i = t * chunk + j;
        unsigned hc = loc[j];
        if (hc && run <= rem && rem < run + hc) {
            selDigit = (unsigned)(NB - 1 - i);
            selBase  = run;
        }
        run += hc;
    }
    __syncthreads();

    if (t == 0) {
        unsigned d    = selDigit;
        unsigned nrem = rem - selBase;
        unsigned npfx = (level == 0) ? d
                      : (level == 1) ? ((pfx << 11) | d)
                                     : ((pfx << 10) | d);   // full 32-bit key
        state[row * 2 + rank] = make_uint2(npfx, nrem);
    }
}

// ---- th0[row] = st + (en - st) * alpha ----------------------------------
__global__ void finalize_th(const uint2* __restrict__ state,
                            const float* __restrict__ alpha,
                            float* __restrict__ th0) {
    int r = threadIdx.x;
    if (r < NROW) {
        float en = ord2f(state[r * 2 + 0].x);
        float st = ord2f(state[r * 2 + 1].x);
        th0[r] = st + (en - st) * alpha[0];
    }
}

// ---- elementwise gate: relu(x) * sigmoid(tau_m * (|x| - th)) ------------
__global__ void apply_gate(const float4* __restrict__ x, const float* __restrict__ th0,
                           const float* __restrict__ tau, float4* __restrict__ out) {
    size_t p = (size_t)blockIdx.x * blockDim.x + threadIdx.x;   // pixel index
    int b = (int)(p >> 18);                                     // p / 262144
    __builtin_prefetch(x + p + 4096, 0, 0);                     // global_prefetch_b8
    float4 v   = x[p];                                          // global_load_b128
    float4 th4 = ((const float4*)th0)[b];                       // th per (b, c0..c3)
    float tv   = tau[0];

    const float xv[4]  = {v.x, v.y, v.z, v.w};
    const float t0v[4] = {th4.x, th4.y, th4.z, th4.w};
    float ov[4];
    #pragma unroll
    for (int c = 0; c < 4; ++c) {
        float t0     = t0v[c];
        float val0   = (t0 > 1e-14f) ? 1.0f : 0.0f;
        float th     = t0 * val0;
        float val_st = th + (1.0f - val0);
        float tau_m  = tv / val_st;
        float z      = tau_m * (fabsf(xv[c]) - th);
        float sg     = 1.0f / (1.0f + __expf(-z));              // v_exp_f32
        ov[c] = fmaxf(xv[c], 0.0f) * sg;
    }
    float4 o; o.x = ov[0]; o.y = ov[1]; o.z = ov[2]; o.w = ov[3];
    out[p] = o;                                                 // global_store_b128
}

extern "C" void kernel_launch(void* const* d_in, const int* in_sizes, int n_in,
                              void* d_out, int out_size, void* d_ws, size_t ws_size,
                              hipStream_t stream) {
    (void)in_sizes; (void)n_in; (void)out_size; (void)ws_size;

    const float* x     = (const float*)d_in[0];
    const float* alpha = (const float*)d_in[1];
    const float* tau   = (const float*)d_in[2];

    // workspace layout: [0, 2MiB) histograms | +2MiB state uint2[128][2] | +2MiB+4KiB th0[128]
    unsigned* hist = (unsigned*)d_ws;
    uint2*    state = (uint2*)((char*)d_ws + (size_t)(2u << 20));
    float*    th0   = (float*)((char*)d_ws + (size_t)(2u << 20) + 4096);

    const int histWordsL01 = NROW * 2 * 2048;   // 524288 (2 MiB)
    const int histWordsL2  = NROW * 2 * 1024;   // 262144 (1 MiB)
    const dim3 hgrid(SEG, B_);

    // level 0: top 11 bits
    zero_ws<<<(histWordsL01 + 255) / 256, 256, 0, stream>>>(hist, histWordsL01);
    hist_l0<<<hgrid, TPB, 0, stream>>>((const float4*)x, hist);
    select_lvl<<<NROW * 2, TPB, 0, stream>>>(hist, state, 0);

    // level 1: next 11 bits
    zero_ws<<<(histWordsL01 + 255) / 256, 256, 0, stream>>>(hist, histWordsL01);
    hist_l1<<<hgrid, TPB, 0, stream>>>((const float4*)x, state, hist);
    select_lvl<<<NROW * 2, TPB, 0, stream>>>(hist, state, 1);

    // level 2: final 10 bits -> exact keys
    zero_ws<<<(histWordsL2 + 255) / 256, 256, 0, stream>>>(hist, histWordsL2);
    hist_l2<<<hgrid, TPB, 0, stream>>>((const float4*)x, state, hist);
    select_lvl<<<NROW * 2, TPB, 0, stream>>>(hist, state, 2);

    finalize_th<<<1, 128, 0, stream>>>(state, alpha, th0);

    apply_gate<<<(B_ * L_) / TPB, TPB, 0, stream>>>((const float4*)x, th0, tau,
                                                    (float4*)d_out);
}